// CausalTrilinearBCNAttention_82360292868508
// MI455X (gfx1250) — compile-verified
//
#include <hip/hip_runtime.h>
#include <hip/hip_bf16.h>
#include <stdint.h>

typedef float v2f __attribute__((ext_vector_type(2)));
typedef float v4f __attribute__((ext_vector_type(4)));
typedef float v8f __attribute__((ext_vector_type(8)));

#define BM 128
#define BN 128
#define BK 16
#define BKP 20   // padded LDS pitch: 16B-aligned rows, conflict-free frag reads

// ---------------------------------------------------------------------------
// C[M,N] = A[M,K] * B[N,K]^T  (+ Cin) (+ bias1 + alpha*bias2), all f32.
// A, B row-major with K contiguous (NT layout). M%128==0, N%128==0, K%16==0.
// V_WMMA_F32_16X16X4_F32, wave tile 64x32 (4x2 frags), block tile 128x128.
// Global->LDS staging uses CDNA5 GLOBAL_LOAD_ASYNC_TO_LDS_B128 (ASYNCcnt).
// ---------------------------------------------------------------------------
__global__ __launch_bounds__(256) void gemm_nt_f32(
    const float* __restrict__ A, const float* __restrict__ B,
    const float* __restrict__ Cin, float* __restrict__ C,
    int M, int N, int Kd,
    const float* __restrict__ bias1, const float* __restrict__ bias2,
    const float* __restrict__ alpha_ptr)
{
  __shared__ float As[2][BM][BKP];
  __shared__ float Bs[2][BN][BKP];

  const int tid  = threadIdx.x;
  const int lane = tid & 31;
  const int wave = tid >> 5;
  const int wm   = wave >> 2;   // 0..1 -> 64-row slab
  const int wn   = wave & 3;    // 0..3 -> 32-col slab
  const int lrow = lane & 15;   // A: row in frag / B: col in frag
  const int lhi  = lane >> 4;   // K-pair select (0 -> K0,1 ; 1 -> K2,3)

  const int m0 = blockIdx.y * BM;
  const int n0 = blockIdx.x * BN;

  // global->LDS mapping: 4 threads per 16-float row, float4 each, 2 row-groups
  const int lr = tid >> 2;          // 0..63
  const int lc = (tid & 3) << 2;    // 0,4,8,12

  v8f acc[4][2];
  if (Cin) {
#pragma unroll
    for (int mi = 0; mi < 4; ++mi)
#pragma unroll
      for (int ni = 0; ni < 2; ++ni) {
        const int n  = n0 + wn * 32 + ni * 16 + lrow;
        const int mb = m0 + wm * 64 + mi * 16 + 8 * lhi;
#pragma unroll
        for (int v = 0; v < 8; ++v)
          acc[mi][ni][v] = Cin[(size_t)(mb + v) * N + n];
      }
  } else {
#pragma unroll
    for (int mi = 0; mi < 4; ++mi)
#pragma unroll
      for (int ni = 0; ni < 2; ++ni)
#pragma unroll
        for (int v = 0; v < 8; ++v)
          acc[mi][ni][v] = 0.0f;
  }

  const int nkb = Kd / BK;

  // Issue one K-tile's worth of async DMA (4 x b128 per thread) into LDS buf.
  // VDST = LDS byte address (low 32 bits of generic pointer), VADDR = 64b global.
  auto issue_tile = [&](int kb, int buf) {
    const float* Ag = A + (size_t)m0 * Kd + (size_t)kb * BK;
    const float* Bg = B + (size_t)n0 * Kd + (size_t)kb * BK;
#pragma unroll
    for (int it = 0; it < 2; ++it) {
      const int r = lr + it * 64;
      {
        const uint32_t la = (uint32_t)(uintptr_t)&As[buf][r][lc];
        const uint64_t ga = (uint64_t)(uintptr_t)(Ag + (size_t)r * Kd + lc);
        asm volatile("global_load_async_to_lds_b128 %0, %1, off"
                     :: "v"(la), "v"(ga) : "memory");
      }
      {
        const uint32_t lb = (uint32_t)(uintptr_t)&Bs[buf][r][lc];
        const uint64_t gb = (uint64_t)(uintptr_t)(Bg + (size_t)r * Kd + lc);
        asm volatile("global_load_async_to_lds_b128 %0, %1, off"
                     :: "v"(lb), "v"(gb) : "memory");
      }
    }
  };

  // Prologue: DMA tile 0 into buffer 0, wait, sync.
  issue_tile(0, 0);
  asm volatile("s_wait_asynccnt 0x0" ::: "memory");
  __syncthreads();

  int cbuf = 0;
  for (int kb = 0; kb < nkb; ++kb) {
    if (kb + 1 < nkb) issue_tile(kb + 1, cbuf ^ 1);  // DMA overlaps compute

#pragma unroll
    for (int kk = 0; kk < BK; kk += 4) {
      v2f af[4], bf[2];
#pragma unroll
      for (int mi = 0; mi < 4; ++mi)
        af[mi] = *(const v2f*)&As[cbuf][wm * 64 + mi * 16 + lrow][kk + 2 * lhi];
#pragma unroll
      for (int ni = 0; ni < 2; ++ni)
        bf[ni] = *(const v2f*)&Bs[cbuf][wn * 32 + ni * 16 + lrow][kk + 2 * lhi];
#pragma unroll
      for (int mi = 0; mi < 4; ++mi)
#pragma unroll
        for (int ni = 0; ni < 2; ++ni)
          acc[mi][ni] = __builtin_amdgcn_wmma_f32_16x16x4_f32(
              false, af[mi], false, bf[ni], (short)0, acc[mi][ni], false, false);
    }

    asm volatile("s_wait_asynccnt 0x0" ::: "memory");  // next tile landed in LDS
    __syncthreads();
    cbuf ^= 1;
  }

  // Epilogue: optional bias1 + alpha*bias2 (per output column)
  float badd[2] = {0.0f, 0.0f};
  if (bias1) {
    float al = 0.0f;
    if (bias2 && alpha_ptr) al = *alpha_ptr;
#pragma unroll
    for (int ni = 0; ni < 2; ++ni) {
      const int n = n0 + wn * 32 + ni * 16 + lrow;
      float v = bias1[n];
      if (bias2) v += al * bias2[n];
      badd[ni] = v;
    }
  }

#pragma unroll
  for (int mi = 0; mi < 4; ++mi)
#pragma unroll
    for (int ni = 0; ni < 2; ++ni) {
      const int n  = n0 + wn * 32 + ni * 16 + lrow;
      const int mb = m0 + wm * 64 + mi * 16 + 8 * lhi;
#pragma unroll
      for (int v = 0; v < 8; ++v)
        C[(size_t)(mb + v) * N + n] = acc[mi][ni][v] + badd[ni];
    }
}

// ---------------------------------------------------------------------------
// [rows,cols] -> [cols,rows] transpose (rows%32==0, cols%32==0), block (32,8)
// ---------------------------------------------------------------------------
__global__ __launch_bounds__(256) void transpose_f32(
    const float* __restrict__ in, float* __restrict__ out, int rows, int cols)
{
  __shared__ float tile[32][33];
  const int bx = blockIdx.x * 32;  // col base
  const int by = blockIdx.y * 32;  // row base
  const int x = threadIdx.x;
  const int y = threadIdx.y;
#pragma unroll
  for (int i = 0; i < 32; i += 8)
    tile[y + i][x] = in[(size_t)(by + y + i) * cols + (bx + x)];
  __syncthreads();
#pragma unroll
  for (int i = 0; i < 32; i += 8)
    out[(size_t)(bx + y + i) * rows + (by + x)] = tile[x][y + i];
}

// ---------------------------------------------------------------------------
// Segmented scans along T for p_b, p_t, xx ([B,T,R]); NSEG segments of SEGL.
// ---------------------------------------------------------------------------
#define NSEG 32
#define SEGL 64   // T / NSEG with T = 2048

__global__ __launch_bounds__(256) void scan_partial(
    const float* __restrict__ pb, const float* __restrict__ pt,
    const float* __restrict__ xx, float* __restrict__ part,
    int Bn, int T, int R)
{
  const int g = blockIdx.x * blockDim.x + threadIdx.x;  // Bn*NSEG*R threads
  const int r   = g % R;
  const int seg = (g / R) % NSEG;
  const int b   = g / (R * NSEG);
  if (b >= Bn) return;
  float s0 = 0.0f, s1 = 0.0f, s2 = 0.0f;
  const int t0 = seg * SEGL;
  for (int t = t0; t < t0 + SEGL; ++t) {
    const size_t idx = ((size_t)(b * T + t)) * R + r;
    s0 += pb[idx]; s1 += pt[idx]; s2 += xx[idx];
  }
  const size_t stride = (size_t)Bn * R * NSEG;
  const size_t pidx   = ((size_t)(b * R + r)) * NSEG + seg;
  part[pidx] = s0; part[pidx + stride] = s1; part[pidx + 2 * stride] = s2;
}

__global__ __launch_bounds__(256) void scan_offsets(float* __restrict__ part, int ncol)
{
  const int g = blockIdx.x * blockDim.x + threadIdx.x;  // ncol = B*R threads
  if (g >= ncol) return;
  const size_t stride = (size_t)ncol * NSEG;
#pragma unroll
  for (int a = 0; a < 3; ++a) {
    float* p = part + (size_t)a * stride + (size_t)g * NSEG;
    float run = 0.0f;
    for (int s = 0; s < NSEG; ++s) { const float v = p[s]; p[s] = run; run += v; }
  }
}

// In-place: a_b <- a_b * cumsum(p_b) / n ;  a_t <- alpha * a_t * cumsum(xx) * cumsum(p_t) / n^2
__global__ __launch_bounds__(256) void scan_apply(
    float* __restrict__ ab, float* __restrict__ at_,
    const float* __restrict__ pb, const float* __restrict__ pt,
    const float* __restrict__ xx, const float* __restrict__ part,
    const float* __restrict__ alpha_ptr, int Bn, int T, int R)
{
  const int g = blockIdx.x * blockDim.x + threadIdx.x;
  const int r   = g % R;
  const int seg = (g / R) % NSEG;
  const int b   = g / (R * NSEG);
  if (b >= Bn) return;
  const float alpha = *alpha_ptr;
  const size_t stride = (size_t)Bn * R * NSEG;
  const size_t pidx   = ((size_t)(b * R + r)) * NSEG + seg;
  float s0 = part[pidx];
  float s1 = part[pidx + stride];
  float s2 = part[pidx + 2 * stride];
  const int t0 = seg * SEGL;
  for (int t = t0; t < t0 + SEGL; ++t) {
    const size_t idx = ((size_t)(b * T + t)) * R + r;
    s0 += pb[idx]; s1 += pt[idx]; s2 += xx[idx];
    const float inv = 1.0f / (float)(t + 1);
    const float gb = ab[idx] * s0 * inv;
    const float gt = alpha * at_[idx] * s2 * s1 * inv * inv;
    ab[idx] = gb;
    at_[idx] = gt;
  }
}

// ---------------------------------------------------------------------------
extern "C" void kernel_launch(void* const* d_in, const int* in_sizes, int n_in,
                              void* d_out, int out_size, void* d_ws, size_t ws_size,
                              hipStream_t stream) {
  (void)in_sizes; (void)n_in; (void)out_size; (void)ws_size;

  const float* x      = (const float*)d_in[0];
  const float* Wq     = (const float*)d_in[1];
  const float* Wk     = (const float*)d_in[2];
  const float* Wo     = (const float*)d_in[3];
  const float* U_b    = (const float*)d_in[4];
  const float* V_b    = (const float*)d_in[5];
  const float* W_b    = (const float*)d_in[6];
  const float* bias_b = (const float*)d_in[7];
  const float* U_t    = (const float*)d_in[8];
  const float* V_t    = (const float*)d_in[9];
  const float* W_t    = (const float*)d_in[10];
  const float* X_t    = (const float*)d_in[11];
  const float* bias_t = (const float*)d_in[12];
  const float* alpha  = (const float*)d_in[13];
  float* out = (float*)d_out;

  const int Bn = 4, T = 2048, D = 1024, R = 256;
  const int M = Bn * T;  // 8192

  float* ws = (float*)d_ws;
  size_t off = 0;
  float* Q    = ws + off; off += (size_t)M * D;   // later reused as `pre`
  float* Kk   = ws + off; off += (size_t)M * D;
  float* a_b  = ws + off; off += (size_t)M * R;   // becomes g_b in place
  float* a_t  = ws + off; off += (size_t)M * R;   // becomes g_t in place
  float* p_b  = ws + off; off += (size_t)M * R;
  float* p_t  = ws + off; off += (size_t)M * R;
  float* xx   = ws + off; off += (size_t)M * R;
  float* VbT  = ws + off; off += (size_t)D * R;
  float* WbT  = ws + off; off += (size_t)D * R;
  float* VtT  = ws + off; off += (size_t)D * R;
  float* WtT  = ws + off; off += (size_t)D * R;
  float* XtT  = ws + off; off += (size_t)D * R;
  float* part = ws + off; off += 3ull * Bn * R * NSEG;

  const dim3 blk(256);
  const dim3 tb(32, 8);
  const dim3 tg(R / 32, D / 32);

  // 1) Put the five [D,R] matrices into [R,D] (NT layout for the GEMM)
  transpose_f32<<<tg, tb, 0, stream>>>(V_b, VbT, D, R);
  transpose_f32<<<tg, tb, 0, stream>>>(W_b, WbT, D, R);
  transpose_f32<<<tg, tb, 0, stream>>>(V_t, VtT, D, R);
  transpose_f32<<<tg, tb, 0, stream>>>(W_t, WtT, D, R);
  transpose_f32<<<tg, tb, 0, stream>>>(X_t, XtT, D, R);

  const dim3 gDD(D / BN, M / BM);   // N=1024
  const dim3 gDR(R / BN, M / BM);   // N=256

  // 2) Q = x Wq^T ; K = x Wk^T
  gemm_nt_f32<<<gDD, blk, 0, stream>>>(x, Wq, nullptr, Q,  M, D, D, nullptr, nullptr, nullptr);
  gemm_nt_f32<<<gDD, blk, 0, stream>>>(x, Wk, nullptr, Kk, M, D, D, nullptr, nullptr, nullptr);

  // 3) rank-R projections
  gemm_nt_f32<<<gDR, blk, 0, stream>>>(Q,  VbT, nullptr, a_b, M, R, D, nullptr, nullptr, nullptr);
  gemm_nt_f32<<<gDR, blk, 0, stream>>>(Q,  VtT, nullptr, a_t, M, R, D, nullptr, nullptr, nullptr);
  gemm_nt_f32<<<gDR, blk, 0, stream>>>(Kk, WbT, nullptr, p_b, M, R, D, nullptr, nullptr, nullptr);
  gemm_nt_f32<<<gDR, blk, 0, stream>>>(Kk, WtT, nullptr, p_t, M, R, D, nullptr, nullptr, nullptr);
  gemm_nt_f32<<<gDR, blk, 0, stream>>>(x,  XtT, nullptr, xx,  M, R, D, nullptr, nullptr, nullptr);

  // 4) causal scans + fused elementwise gating (g_b, g_t written in place)
  const int scanThreads = Bn * NSEG * R;  // 32768
  scan_partial<<<scanThreads / 256, blk, 0, stream>>>(p_b, p_t, xx, part, Bn, T, R);
  scan_offsets<<<(Bn * R + 255) / 256, blk, 0, stream>>>(part, Bn * R);
  scan_apply<<<scanThreads / 256, blk, 0, stream>>>(a_b, a_t, p_b, p_t, xx, part, alpha, Bn, T, R);

  // 5) pre = g_b U_b^T + g_t U_t^T + (bias_b + alpha*bias_t)   (reuse Q buffer)
  gemm_nt_f32<<<gDD, blk, 0, stream>>>(a_b, U_b, nullptr, Q, M, D, R, bias_b, bias_t, alpha);
  gemm_nt_f32<<<gDD, blk, 0, stream>>>(a_t, U_t, Q,       Q, M, D, R, nullptr, nullptr, nullptr);

  // 6) out = pre Wo^T
  gemm_nt_f32<<<gDD, blk, 0, stream>>>(Q, Wo, nullptr, out, M, D, D, nullptr, nullptr, nullptr);
}